// BuiltinGNN_77326591197791
// MI455X (gfx1250) — compile-verified
//
#include <hip/hip_runtime.h>
#include <hip/hip_bf16.h>

#define N_NODES 50000
#define N_EDGES 600000
#define HID     128
#define OUT_C   8
#define N_GRAPHS 64
#define NM ((size_t)N_NODES * HID)

typedef __attribute__((ext_vector_type(2))) float v2f;
typedef __attribute__((ext_vector_type(8))) float v8f;

// ---------------------------------------------------------------- utilities
__global__ __launch_bounds__(256) void zero_kernel(float* __restrict__ p, size_t n) {
    size_t i = (size_t)blockIdx.x * 256 + threadIdx.x;
    if (i < n) p[i] = 0.0f;
}

// deg[dst] += 1 per edge
__global__ __launch_bounds__(256) void deg_kernel(const int* __restrict__ dst,
                                                  float* __restrict__ deg, int nE) {
    int e = blockIdx.x * 256 + threadIdx.x;
    if (e < nE) atomicAdd(&deg[dst[e]], 1.0f);
}

// dinv = deg>0 ? rsqrt(deg) : 0   (in place)
__global__ __launch_bounds__(256) void dinv_kernel(float* __restrict__ deg, int n) {
    int i = blockIdx.x * 256 + threadIdx.x;
    if (i < n) {
        float d = deg[i];
        deg[i] = (d > 0.0f) ? rsqrtf(d) : 0.0f;
    }
}

// ew[e] = dinv[src[e]] * dinv[dst[e]]
__global__ __launch_bounds__(256) void edgew_kernel(const int* __restrict__ src,
                                                    const int* __restrict__ dst,
                                                    const float* __restrict__ dinv,
                                                    float* __restrict__ ew, int nE) {
    int e = blockIdx.x * 256 + threadIdx.x;
    if (e < nE) ew[e] = dinv[src[e]] * dinv[dst[e]];
}

// cnt[batch[i]] += 1
__global__ __launch_bounds__(256) void count_kernel(const int* __restrict__ batch,
                                                    float* __restrict__ cnt, int n) {
    int i = blockIdx.x * 256 + threadIdx.x;
    if (i < n) atomicAdd(&cnt[batch[i]], 1.0f);
}

// ------------------------------------------------- fp32 WMMA GEMM: H = X @ W
// X: [M,128], W: [128,128] row-major, H: [M,128]. M % 16 == 0.
// Block = 256 threads = 8 waves. Each wave: one 16-row tile, all 8 N-tiles.
__global__ __launch_bounds__(256) void gemm_wmma_kernel(const float* __restrict__ X,
                                                        const float* __restrict__ W,
                                                        float* __restrict__ H, int M) {
    __shared__ float wlds[HID * HID];   // 64 KB, row-major copy of W
    const int tid = threadIdx.x;
    // coalesced float4 fill: 4096 float4 / 256 threads
    {
        const float4* ws = (const float4*)W;
        float4* wd = (float4*)wlds;
        #pragma unroll
        for (int i = 0; i < 16; ++i) wd[tid + i * 256] = ws[tid + i * 256];
    }
    __syncthreads();

    const int wave  = tid >> 5;
    const int lane  = tid & 31;
    const int tileM = blockIdx.x * 128 + wave * 16;
    if (tileM >= M) return;

    const int mlane = lane & 15;        // row-in-tile (A) / column-in-tile (B,D)
    const int khalf = (lane >> 4) * 2;  // 0 or 2: which K pair this lane holds

    v8f acc[8];
    #pragma unroll
    for (int t = 0; t < 8; ++t)
        acc[t] = (v8f){0.f, 0.f, 0.f, 0.f, 0.f, 0.f, 0.f, 0.f};

    const float* xrow = X + (size_t)(tileM + mlane) * HID;

    #pragma unroll 4
    for (int k0 = 0; k0 < HID; k0 += 4) {
        const int kb = k0 + khalf;                    // even -> 8B aligned
        // A fragment: 16x4 f32 (ISA 7.12.2 layout)
        v2f a = *(const v2f*)(xrow + kb);
        #pragma unroll
        for (int nt = 0; nt < 8; ++nt) {
            const int nc = nt * 16 + mlane;
            // B fragment: 4x16 f32; lanes 0-15 -> banks 0..15 (conflict-free)
            v2f b;
            b.x = wlds[kb * HID + nc];
            b.y = wlds[(kb + 1) * HID + nc];
            acc[nt] = __builtin_amdgcn_wmma_f32_16x16x4_f32(
                false, a, false, b, (short)0, acc[nt], false, false);
        }
    }

    // D layout: VGPR r -> (lanes 0-15: M=r, N=lane) (lanes 16-31: M=r+8, N=lane-16)
    const int rbase = tileM + (lane >> 4) * 8;
    #pragma unroll
    for (int nt = 0; nt < 8; ++nt) {
        #pragma unroll
        for (int r = 0; r < 8; ++r)
            H[(size_t)(rbase + r) * HID + nt * 16 + mlane] = acc[nt][r];
    }
}

// ------------------------------------------ edge scatter: AGG[dst] += w*H[src]
// 32 lanes per edge, float4 (4 channels) per lane.
__global__ __launch_bounds__(256) void scatter_kernel(const float* __restrict__ H,
                                                      const int* __restrict__ src,
                                                      const int* __restrict__ dst,
                                                      const float* __restrict__ ew,
                                                      float* __restrict__ AGG, int nE) {
    int e = blockIdx.x * 8 + (threadIdx.x >> 5);
    if (e >= nE) return;
    const int lane = threadIdx.x & 31;
    const int s = src[e], d = dst[e];
    const float w = ew[e];
    const float4 hv = *(const float4*)(H + (size_t)s * HID + lane * 4);
    float* ap = AGG + (size_t)d * HID + lane * 4;
    atomicAdd(ap + 0, w * hv.x);
    atomicAdd(ap + 1, w * hv.y);
    atomicAdd(ap + 2, w * hv.z);
    atomicAdd(ap + 3, w * hv.w);
}

// -------------------------------------------------- BatchNorm statistics pass
#define STAT_ROWS 256
__global__ __launch_bounds__(128) void stats_kernel(const float* __restrict__ AGG,
                                                    const float* __restrict__ bias,
                                                    float* __restrict__ stats, int M) {
    const int c = threadIdx.x;
    const float b = bias[c];
    int r0 = blockIdx.x * STAT_ROWS;
    int r1 = r0 + STAT_ROWS; if (r1 > M) r1 = M;
    float s = 0.f, ss = 0.f;
    for (int r = r0; r < r1; ++r) {
        float v = AGG[(size_t)r * HID + c] + b;
        s += v; ss += v * v;
    }
    atomicAdd(&stats[c], s);
    atomicAdd(&stats[HID + c], ss);
}

// stats -> (scale, shift):  y = agg*scale + shift
__global__ __launch_bounds__(128) void bn_finalize_kernel(float* __restrict__ stats,
                                                          const float* __restrict__ g,
                                                          const float* __restrict__ be,
                                                          const float* __restrict__ bias,
                                                          int M) {
    const int c = threadIdx.x;
    float mean = stats[c] / (float)M;
    float var  = stats[HID + c] / (float)M - mean * mean;
    float rstd = rsqrtf(var + 1e-5f);
    float scale = g[c] * rstd;
    stats[c]       = scale;
    stats[HID + c] = (bias[c] - mean) * scale + be[c];
}

__global__ __launch_bounds__(256) void bn_relu_kernel(const float* __restrict__ AGG,
                                                      const float* __restrict__ stats,
                                                      float* __restrict__ Xo, size_t n) {
    size_t i = (size_t)blockIdx.x * 256 + threadIdx.x;
    if (i >= n) return;
    int c = (int)(i & (HID - 1));
    float y = AGG[i] * stats[c] + stats[HID + c];
    Xo[i] = fmaxf(y, 0.0f);
}

// layer-3: BN+ReLU fused with per-graph pooling sum (batch is sorted ->
// local accumulate, flush on graph change)
__global__ __launch_bounds__(128) void bn_relu_pool_kernel(const float* __restrict__ AGG,
                                                           const float* __restrict__ stats,
                                                           const int* __restrict__ batch,
                                                           float* __restrict__ pooled, int M) {
    const int c = threadIdx.x;
    const float sc = stats[c], sh = stats[HID + c];
    int r0 = blockIdx.x * STAT_ROWS;
    int r1 = r0 + STAT_ROWS; if (r1 > M) r1 = M;
    float acc = 0.f; int curg = -1;
    for (int r = r0; r < r1; ++r) {
        int gidx = batch[r];
        if (gidx != curg) {
            if (curg >= 0) atomicAdd(&pooled[(size_t)curg * HID + c], acc);
            curg = gidx; acc = 0.f;
        }
        acc += fmaxf(AGG[(size_t)r * HID + c] * sc + sh, 0.0f);
    }
    if (curg >= 0) atomicAdd(&pooled[(size_t)curg * HID + c], acc);
}

// out[g][oc] = (pooled[g]/cnt[g]) @ Wh + bh   (64x8 outputs, one block)
__global__ __launch_bounds__(512) void head_kernel(const float* __restrict__ pooled,
                                                   const float* __restrict__ cnt,
                                                   const float* __restrict__ Wh,
                                                   const float* __restrict__ bh,
                                                   float* __restrict__ out) {
    const int t = threadIdx.x;            // 512 = 64 graphs * 8 outputs
    const int g = t >> 3, oc = t & 7;
    float inv = 1.0f / fmaxf(cnt[g], 1.0f);
    float s = 0.0f;
    #pragma unroll 8
    for (int k = 0; k < HID; ++k)
        s += pooled[(size_t)g * HID + k] * Wh[k * OUT_C + oc];
    out[t] = s * inv + bh[oc];
}

// ---------------------------------------------------------------- launcher
extern "C" void kernel_launch(void* const* d_in, const int* in_sizes, int n_in,
                              void* d_out, int out_size, void* d_ws, size_t ws_size,
                              hipStream_t stream) {
    const float* X0    = (const float*)d_in[0];
    const int*   EI    = (const int*)d_in[1];
    const int*   BATCH = (const int*)d_in[2];
    const float* Wl[3]  = {(const float*)d_in[3],  (const float*)d_in[7],  (const float*)d_in[11]};
    const float* bl[3]  = {(const float*)d_in[4],  (const float*)d_in[8],  (const float*)d_in[12]};
    const float* gl[3]  = {(const float*)d_in[5],  (const float*)d_in[9],  (const float*)d_in[13]};
    const float* bel[3] = {(const float*)d_in[6],  (const float*)d_in[10], (const float*)d_in[14]};
    const float* Wh = (const float*)d_in[15];
    const float* bh = (const float*)d_in[16];
    float* out = (float*)d_out;

    const int* SRC = EI;
    const int* DST = EI + N_EDGES;

    // workspace carve-up (floats)
    float* ws     = (float*)d_ws;
    float* XBUF   = ws;                       // NM
    float* HBUF   = XBUF + NM;                // NM
    float* AGG    = HBUF + NM;                // NM
    float* DINV   = AGG + NM;                 // N_NODES (deg -> dinv in place)
    float* EW     = DINV + N_NODES;           // N_EDGES
    float* STATS  = EW + N_EDGES;             // 2*HID
    float* POOLED = STATS + 2 * HID;          // N_GRAPHS*HID
    float* CNT    = POOLED + N_GRAPHS * HID;  // N_GRAPHS

    const int nodeBlk = (N_NODES + 255) / 256;
    const int edgeBlk = (N_EDGES + 255) / 256;
    const int nmBlk   = (int)((NM + 255) / 256);
    const int statBlk = (N_NODES + STAT_ROWS - 1) / STAT_ROWS;
    const int gemmBlk = (N_NODES + 127) / 128;

    // ---- degree / edge weights / pool counts (shared across layers)
    zero_kernel<<<nodeBlk, 256, 0, stream>>>(DINV, N_NODES);
    zero_kernel<<<(N_GRAPHS * HID + N_GRAPHS + 255) / 256, 256, 0, stream>>>(POOLED,
                                                         N_GRAPHS * HID + N_GRAPHS);
    deg_kernel<<<edgeBlk, 256, 0, stream>>>(DST, DINV, N_EDGES);
    dinv_kernel<<<nodeBlk, 256, 0, stream>>>(DINV, N_NODES);
    edgew_kernel<<<edgeBlk, 256, 0, stream>>>(SRC, DST, DINV, EW, N_EDGES);
    count_kernel<<<nodeBlk, 256, 0, stream>>>(BATCH, CNT, N_NODES);

    const float* Xcur = X0;
    for (int l = 0; l < 3; ++l) {
        // H = Xcur @ W_l  (V_WMMA_F32_16X16X4_F32)
        gemm_wmma_kernel<<<gemmBlk, 256, 0, stream>>>(Xcur, Wl[l], HBUF, N_NODES);
        // AGG = scatter-add of normalized messages
        zero_kernel<<<nmBlk, 256, 0, stream>>>(AGG, NM);
        scatter_kernel<<<(N_EDGES + 7) / 8, 256, 0, stream>>>(HBUF, SRC, DST, EW, AGG, N_EDGES);
        // BatchNorm stats on (AGG + b)
        zero_kernel<<<1, 256, 0, stream>>>(STATS, 2 * HID);
        stats_kernel<<<statBlk, 128, 0, stream>>>(AGG, bl[l], STATS, N_NODES);
        bn_finalize_kernel<<<1, 128, 0, stream>>>(STATS, gl[l], bel[l], bl[l], N_NODES);
        if (l < 2) {
            bn_relu_kernel<<<nmBlk, 256, 0, stream>>>(AGG, STATS, XBUF, NM);
            Xcur = XBUF;
        } else {
            bn_relu_pool_kernel<<<statBlk, 128, 0, stream>>>(AGG, STATS, BATCH, POOLED, N_NODES);
        }
    }

    head_kernel<<<1, 512, 0, stream>>>(POOLED, CNT, Wh, bh, out);
}